// Model_77008763617996
// MI455X (gfx1250) — compile-verified
//
#include <hip/hip_runtime.h>

// Problem constants (match reference)
#define BB    16
#define NN    20000
#define EE    640000
#define NREGC 2000
#define DINC  128
#define FFC   32
#define NEGSLOPE 0.2f

typedef float v2f __attribute__((ext_vector_type(2)));
typedef float v8f __attribute__((ext_vector_type(8)));
typedef unsigned int v4u __attribute__((ext_vector_type(4)));
typedef int v4i __attribute__((ext_vector_type(4)));
typedef int v8i __attribute__((ext_vector_type(8)));

#if __has_builtin(__builtin_amdgcn_tensor_load_to_lds) && __has_builtin(__builtin_amdgcn_s_wait_tensorcnt)
#define USE_TDM 1
#endif

__device__ __forceinline__ float lrelu(float x) { return x > 0.f ? x : NEGSLOPE * x; }

// ---------------------------------------------------------------------------
// 0) init: zero per-batch edge_attr sums; compute c_edge = dot(lin_edge, att_edge)
// ---------------------------------------------------------------------------
__global__ void k_init(const float* __restrict__ lin_e, const float* __restrict__ att_e,
                       float* __restrict__ msum, float* __restrict__ cedge) {
  int t = threadIdx.x;
  if (t < BB) msum[t] = 0.f;
  if (t == 0) {
    float c = 0.f;
    for (int f = 0; f < FFC; ++f) c += lin_e[f] * att_e[f];
    cedge[0] = c;
  }
}

// ---------------------------------------------------------------------------
// 1) per-batch sum of edge_attr (for self-loop fill_value='mean')
// ---------------------------------------------------------------------------
__global__ void k_mean(const float* __restrict__ eattr, float* __restrict__ msum) {
  __shared__ float sd[256];
  int b     = blockIdx.x / (EE / 256);
  int chunk = blockIdx.x % (EE / 256);
  int e     = chunk * 256 + threadIdx.x;
  sd[threadIdx.x] = eattr[(size_t)b * EE + e];
  __syncthreads();
  for (int s = 128; s > 0; s >>= 1) {
    if (threadIdx.x < s) sd[threadIdx.x] += sd[threadIdx.x + s];
    __syncthreads();
  }
  if (threadIdx.x == 0) atomicAdd(&msum[b], sd[0]);
}

// ---------------------------------------------------------------------------
// 2) h = x @ W via fp32 WMMA 16x16x4. 8 waves/block; W staged in LDS via the
//    Tensor Data Mover (one 16KB contiguous tile), B-frags served by ds_load.
//    A frag (16x4 f32): lane m=lane&15, k-pair base kb=(lane>>4)*2
//    B frag (4x16 f32): lane n=lane&15, same kb
//    D (16x16 f32, 8 VGPRs): row = v + 8*(lane>>4), col = lane&15
// ---------------------------------------------------------------------------
__global__ void __launch_bounds__(256) k_gemm(const float* __restrict__ x,
                                              const float* __restrict__ W,
                                              float* __restrict__ h) {
  __shared__ float shW[DINC * FFC];  // 16 KB

#ifdef USE_TDM
  if (threadIdx.x == 0) {
    unsigned long long ga = (unsigned long long)W;
    unsigned lds_off = (unsigned)(unsigned long long)(void*)&shW[0];
    // D# group0: count=1 | lds_addr | global_addr[56:0] | type=2
    v4u g0 = { 1u,
               lds_off,
               (unsigned)(ga & 0xFFFFFFFFull),
               ((unsigned)((ga >> 32) & 0x01FFFFFFull)) | 0x80000000u };
    // D# group1: data_size=2(4B); tensor_dim0=4096; tensor_dim1=1;
    //            tile_dim0=4096; tile_dim1=1; dim0_stride=4096; dim1_stride=4096
    v8i g1 = { (int)0x00020000u,          // wg_mask=0, data_size=4B, no flags
               (int)(4096u << 16),        // [15:0] bar_addr=0 ; [31:16] tensor_dim0.lo
               (int)(1u << 16),           // [15:0] tensor_dim0.hi=0 ; [31:16] tensor_dim1.lo=1
               (int)(4096u << 16),        // [15:0] tensor_dim1.hi=0 ; [31:16] tile_dim0=4096
               (int)1,                    // [15:0] tile_dim1=1 ; [31:16] tile_dim2=0
               (int)4096,                 // tensor_dim0_stride[31:0]
               (int)(4096u << 16),        // [15:0] stride0.hi=0 ; [31:16] dim1_stride.lo
               0 };                       // dim1_stride hi
    v4i gz = { 0, 0, 0, 0 };
#if __has_include(<hip/amd_detail/amd_gfx1250_TDM.h>)
    v8i g4 = { 0, 0, 0, 0, 0, 0, 0, 0 }; // 6-arg (therock) form
    __builtin_amdgcn_tensor_load_to_lds(g0, g1, gz, gz, g4, 0);
#else
    __builtin_amdgcn_tensor_load_to_lds(g0, g1, gz, gz, 0); // 5-arg (ROCm 7.2)
#endif
    __builtin_amdgcn_s_wait_tensorcnt(0);
  }
  __syncthreads();
#else
  for (int i = threadIdx.x; i < (DINC * FFC) / 4; i += 256)
    ((float4*)shW)[i] = ((const float4*)W)[i];
  __syncthreads();
#endif

  int wave = blockIdx.x * 8 + (threadIdx.x >> 5);  // exactly BB*(NN/16) waves
  int lane = threadIdx.x & 31;
  int b  = wave / (NN / 16);
  int tr = (wave % (NN / 16)) * 16;
  int m  = lane & 15;
  int kb = (lane >> 4) << 1;
  const float* xrow = x + ((size_t)b * NN + tr + m) * DINC;
  v8f acc0 = {0.f, 0.f, 0.f, 0.f, 0.f, 0.f, 0.f, 0.f};
  v8f acc1 = {0.f, 0.f, 0.f, 0.f, 0.f, 0.f, 0.f, 0.f};
  for (int k0 = 0; k0 < DINC; k0 += 4) {
    v2f a, b0, b1;
    a.x  = xrow[k0 + kb];
    a.y  = xrow[k0 + kb + 1];
    b0.x = shW[(k0 + kb) * FFC + m];
    b0.y = shW[(k0 + kb + 1) * FFC + m];
    b1.x = shW[(k0 + kb) * FFC + 16 + m];
    b1.y = shW[(k0 + kb + 1) * FFC + 16 + m];
    acc0 = __builtin_amdgcn_wmma_f32_16x16x4_f32(false, a, false, b0, (short)0, acc0, false, false);
    acc1 = __builtin_amdgcn_wmma_f32_16x16x4_f32(false, a, false, b1, (short)0, acc1, false, false);
  }
  float* hb = h + ((size_t)b * NN + tr) * FFC;
  int rbase = (lane >> 4) << 3;
  for (int v = 0; v < 8; ++v) {
    hb[(size_t)(rbase + v) * FFC + m]      = acc0[v];
    hb[(size_t)(rbase + v) * FFC + 16 + m] = acc1[v];
  }
}

// ---------------------------------------------------------------------------
// 3) a_src[i] = h[i,:]·att_src ; a_dst[i] = h[i,:]·att_dst  (i over B*N)
// ---------------------------------------------------------------------------
__global__ void k_adot(const float* __restrict__ h, const float* __restrict__ as,
                       const float* __restrict__ ad, float* __restrict__ asrc,
                       float* __restrict__ adst) {
  int t = blockIdx.x * blockDim.x + threadIdx.x;
  if (t >= BB * NN) return;
  const float* hr = h + (size_t)t * FFC;
  float s1 = 0.f, s2 = 0.f;
  for (int f = 0; f < FFC; ++f) {
    float v = hr[f];
    s1 += v * as[f];
    s2 += v * ad[f];
  }
  asrc[t] = s1;
  adst[t] = s2;
}

// ---------------------------------------------------------------------------
// 4) m init with self-loop alpha (fill_value='mean' edge_attr)
// ---------------------------------------------------------------------------
__global__ void k_selfmax(const float* __restrict__ asrc, const float* __restrict__ adst,
                          const float* __restrict__ msum, const float* __restrict__ cedge,
                          float* __restrict__ mmax) {
  int t = blockIdx.x * blockDim.x + threadIdx.x;
  if (t >= BB * NN) return;
  int b = t / NN;
  float mean = msum[b] * (1.0f / (float)EE);
  mmax[t] = lrelu(asrc[t] + adst[t] + cedge[0] * mean);
}

// ---------------------------------------------------------------------------
// 5) segment max over edges (atomic f32 max)
// ---------------------------------------------------------------------------
__global__ void k_edge_max(const long long* __restrict__ src, const long long* __restrict__ dst,
                           const float* __restrict__ eattr, const float* __restrict__ asrc,
                           const float* __restrict__ adst, const float* __restrict__ cedge,
                           float* __restrict__ mmax) {
  long long t = (long long)blockIdx.x * blockDim.x + threadIdx.x;
  if (t >= (long long)BB * EE) return;
  int b = (int)(t / EE), e = (int)(t % EE);
  int s = (int)src[e], d = (int)dst[e];
  float al = lrelu(asrc[b * NN + s] + adst[b * NN + d] + cedge[0] * eattr[(size_t)b * EE + e]);
  atomicMax(&mmax[b * NN + d], al);
}

// ---------------------------------------------------------------------------
// 6) s init with self-loop exp
// ---------------------------------------------------------------------------
__global__ void k_selfsum(const float* __restrict__ asrc, const float* __restrict__ adst,
                          const float* __restrict__ msum, const float* __restrict__ cedge,
                          const float* __restrict__ mmax, float* __restrict__ ssum) {
  int t = blockIdx.x * blockDim.x + threadIdx.x;
  if (t >= BB * NN) return;
  int b = t / NN;
  float al = lrelu(asrc[t] + adst[t] + cedge[0] * msum[b] * (1.0f / (float)EE));
  ssum[t] = __expf(al - mmax[t]);
}

// ---------------------------------------------------------------------------
// 7) segment sum of exp over edges
// ---------------------------------------------------------------------------
__global__ void k_edge_sum(const long long* __restrict__ src, const long long* __restrict__ dst,
                           const float* __restrict__ eattr, const float* __restrict__ asrc,
                           const float* __restrict__ adst, const float* __restrict__ cedge,
                           const float* __restrict__ mmax, float* __restrict__ ssum) {
  long long t = (long long)blockIdx.x * blockDim.x + threadIdx.x;
  if (t >= (long long)BB * EE) return;
  int b = (int)(t / EE), e = (int)(t % EE);
  int s = (int)src[e], d = (int)dst[e];
  float al = lrelu(asrc[b * NN + s] + adst[b * NN + d] + cedge[0] * eattr[(size_t)b * EE + e]);
  atomicAdd(&ssum[b * NN + d], __expf(al - mmax[b * NN + d]));
}

// ---------------------------------------------------------------------------
// 8) rep init: self-loop coef * h + bias
// ---------------------------------------------------------------------------
__global__ void k_selfrep(const float* __restrict__ h, const float* __restrict__ asrc,
                          const float* __restrict__ adst, const float* __restrict__ msum,
                          const float* __restrict__ cedge, const float* __restrict__ mmax,
                          const float* __restrict__ ssum, const float* __restrict__ bias,
                          float* __restrict__ rep) {
  long long t = (long long)blockIdx.x * blockDim.x + threadIdx.x;
  if (t >= (long long)BB * NN * FFC) return;
  int i = (int)(t / FFC);
  int f = (int)(t % FFC);
  int b = i / NN;
  float al   = lrelu(asrc[i] + adst[i] + cedge[0] * msum[b] * (1.0f / (float)EE));
  float coef = __expf(al - mmax[i]) / ssum[i];
  rep[t] = coef * h[t] + bias[f];
}

// ---------------------------------------------------------------------------
// 9) edge scatter: rep[dst,:] += coef_e * h[src,:]  (one wave per edge, lane=f)
// ---------------------------------------------------------------------------
__global__ void k_edge_rep(const long long* __restrict__ src, const long long* __restrict__ dst,
                           const float* __restrict__ eattr, const float* __restrict__ asrc,
                           const float* __restrict__ adst, const float* __restrict__ cedge,
                           const float* __restrict__ mmax, const float* __restrict__ ssum,
                           const float* __restrict__ h, float* __restrict__ rep) {
  long long wid = ((long long)blockIdx.x * blockDim.x + threadIdx.x) >> 5;
  int lane = threadIdx.x & 31;
  if (wid >= (long long)BB * EE) return;
  int b = (int)(wid / EE), e = (int)(wid % EE);
  int s = (int)src[e], d = (int)dst[e];
  float al   = lrelu(asrc[b * NN + s] + adst[b * NN + d] + cedge[0] * eattr[(size_t)b * EE + e]);
  float coef = __expf(al - mmax[b * NN + d]) / ssum[b * NN + d];
  float val  = coef * h[((size_t)b * NN + s) * FFC + lane];
  atomicAdd(&rep[((size_t)b * NN + d) * FFC + lane], val);
}

// ---------------------------------------------------------------------------
// 10) out init = rep[:NREG]
// ---------------------------------------------------------------------------
__global__ void k_outinit(const float* __restrict__ rep, float* __restrict__ out) {
  int t = blockIdx.x * blockDim.x + threadIdx.x;
  if (t >= BB * NREGC * FFC) return;
  int b   = t / (NREGC * FFC);
  int rem = t % (NREGC * FFC);
  int r   = rem / FFC;
  int f   = rem % FFC;
  out[t] = rep[((size_t)b * NN + r) * FFC + f];
}

// ---------------------------------------------------------------------------
// 11) regulon pooling: out[src,:] += rep[dst,:]  (one wave per edge, lane=f)
// ---------------------------------------------------------------------------
__global__ void k_edge_pool(const long long* __restrict__ src, const long long* __restrict__ dst,
                            const float* __restrict__ rep, float* __restrict__ out) {
  long long wid = ((long long)blockIdx.x * blockDim.x + threadIdx.x) >> 5;
  int lane = threadIdx.x & 31;
  if (wid >= (long long)BB * EE) return;
  int b = (int)(wid / EE), e = (int)(wid % EE);
  int s = (int)src[e], d = (int)dst[e];
  float v = rep[((size_t)b * NN + d) * FFC + lane];
  atomicAdd(&out[((size_t)b * NREGC + s) * FFC + lane], v);
}

// ---------------------------------------------------------------------------
extern "C" void kernel_launch(void* const* d_in, const int* in_sizes, int n_in,
                              void* d_out, int out_size, void* d_ws, size_t ws_size,
                              hipStream_t stream) {
  const float*     x     = (const float*)d_in[0];
  const float*     eattr = (const float*)d_in[1];
  const float*     W     = (const float*)d_in[2];
  const float*     a_src = (const float*)d_in[3];
  const float*     a_dst = (const float*)d_in[4];
  const float*     lin_e = (const float*)d_in[5];
  const float*     att_e = (const float*)d_in[6];
  const float*     bias  = (const float*)d_in[7];
  const long long* esrc  = (const long long*)d_in[8];
  const long long* edst  = (const long long*)d_in[9];
  float* out = (float*)d_out;

  float* ws   = (float*)d_ws;
  float* h    = ws;                           // B*N*F
  float* rep  = h    + (size_t)BB * NN * FFC; // B*N*F
  float* asrc = rep  + (size_t)BB * NN * FFC; // B*N
  float* adst = asrc + (size_t)BB * NN;       // B*N
  float* mmax = adst + (size_t)BB * NN;       // B*N
  float* ssum = mmax + (size_t)BB * NN;       // B*N
  float* msum = ssum + (size_t)BB * NN;       // B
  float* cedg = msum + BB;                    // 1

  const int nodeBlocks = (BB * NN + 255) / 256;
  const int edgeBlocks = (int)(((long long)BB * EE + 255) / 256);
  const int edgeWaveBlocks = (int)(((long long)BB * EE * 32 + 255) / 256);

  k_init<<<1, 64, 0, stream>>>(lin_e, att_e, msum, cedg);
  k_mean<<<BB * (EE / 256), 256, 0, stream>>>(eattr, msum);
  k_gemm<<<(BB * (NN / 16)) / 8, 256, 0, stream>>>(x, W, h);
  k_adot<<<nodeBlocks, 256, 0, stream>>>(h, a_src, a_dst, asrc, adst);
  k_selfmax<<<nodeBlocks, 256, 0, stream>>>(asrc, adst, msum, cedg, mmax);
  k_edge_max<<<edgeBlocks, 256, 0, stream>>>(esrc, edst, eattr, asrc, adst, cedg, mmax);
  k_selfsum<<<nodeBlocks, 256, 0, stream>>>(asrc, adst, msum, cedg, mmax, ssum);
  k_edge_sum<<<edgeBlocks, 256, 0, stream>>>(esrc, edst, eattr, asrc, adst, cedg, mmax, ssum);
  k_selfrep<<<(int)(((long long)BB * NN * FFC + 255) / 256), 256, 0, stream>>>(
      h, asrc, adst, msum, cedg, mmax, ssum, bias, rep);
  k_edge_rep<<<edgeWaveBlocks, 256, 0, stream>>>(esrc, edst, eattr, asrc, adst, cedg,
                                                 mmax, ssum, h, rep);
  k_outinit<<<(BB * NREGC * FFC + 255) / 256, 256, 0, stream>>>(rep, out);
  k_edge_pool<<<edgeWaveBlocks, 256, 0, stream>>>(esrc, edst, rep, out);
}